// SimpleGAT_4638564679934
// MI455X (gfx1250) — compile-verified
//
#include <hip/hip_runtime.h>
#include <hip/hip_bf16.h>
#include <stdint.h>

#define NODES  100000
#define EDGES  1600000
#define TOTE   (EDGES + NODES)   // edges + self loops
#define GRAPHS 2048
#define DOUT   64                // HEADS*HID = 4*16

typedef __attribute__((ext_vector_type(2))) float v2f;
typedef __attribute__((ext_vector_type(8))) float v8f;

// ---- monotone float<->uint key for atomic max on floats -------------------
__device__ __forceinline__ unsigned fkey(float f) {
  unsigned b = __float_as_uint(f);
  return (b & 0x80000000u) ? ~b : (b | 0x80000000u);
}
__device__ __forceinline__ float funkey(unsigned k) {
  unsigned b = (k & 0x80000000u) ? (k & 0x7FFFFFFFu) : ~k;
  return __uint_as_float(b);
}

// ---- H = X @ W, X:[N,FIN] f32, W:[FIN,64] f32, H:[N,64] f32 ---------------
// One wave computes a 16x64 tile via 4 accumulators of V_WMMA_F32_16X16X4_F32.
// W staged in LDS. 8 waves / block -> 128 rows per block.
template <int FIN>
__global__ __launch_bounds__(256) void gat_gemm_wmma(
    const float* __restrict__ X, const float* __restrict__ W,
    float* __restrict__ H) {
  __shared__ float sW[FIN * DOUT];
  for (int i = threadIdx.x; i < FIN * DOUT; i += 256) sW[i] = W[i];
  __syncthreads();

  const int lane = threadIdx.x & 31;
  const int wave = threadIdx.x >> 5;
  const int m0 = (blockIdx.x * 8 + wave) * 16;
  if (m0 >= NODES) return;

  // A fragment (16x4 f32): lanes 0-15 hold K={0,1}, lanes 16-31 hold K={2,3}
  const int mrow = lane & 15;
  const int kh   = (lane >> 4) << 1;   // 0 or 2
  const int ncol = lane & 15;          // B/C column within 16-wide tile

  v8f acc0 = {}, acc1 = {}, acc2 = {}, acc3 = {};
  const float* __restrict__ xp = X + (size_t)(m0 + mrow) * FIN;

  for (int k = 0; k < FIN; k += 4) {
    v2f a;
    a.x = xp[k + kh];
    a.y = xp[k + kh + 1];
    // B fragment (4x16 f32): row (k+kh) in v.x, row (k+kh+1) in v.y
    const float* __restrict__ w0 = &sW[(k + kh) * DOUT + ncol];
    const float* __restrict__ w1 = w0 + DOUT;
    v2f b;
    b.x = w0[0];  b.y = w1[0];
    acc0 = __builtin_amdgcn_wmma_f32_16x16x4_f32(false, a, false, b, (short)0, acc0, false, false);
    b.x = w0[16]; b.y = w1[16];
    acc1 = __builtin_amdgcn_wmma_f32_16x16x4_f32(false, a, false, b, (short)0, acc1, false, false);
    b.x = w0[32]; b.y = w1[32];
    acc2 = __builtin_amdgcn_wmma_f32_16x16x4_f32(false, a, false, b, (short)0, acc2, false, false);
    b.x = w0[48]; b.y = w1[48];
    acc3 = __builtin_amdgcn_wmma_f32_16x16x4_f32(false, a, false, b, (short)0, acc3, false, false);
  }

  // C/D 16x16 f32 layout: VGPR r -> M = r (lanes 0-15) / r+8 (lanes 16-31)
  const int rbase = m0 + ((lane >> 4) << 3);
#pragma unroll
  for (int r = 0; r < 8; ++r) {
    float* hp = H + (size_t)(rbase + r) * DOUT + ncol;
    hp[0]  = acc0[r];
    hp[16] = acc1[r];
    hp[32] = acc2[r];
    hp[48] = acc3[r];
  }
}

// ---- attention logits: al[n,h] = <h[n,h,:], a[h,:]> -----------------------
__global__ void att_logits(const float* __restrict__ H,
                           const float* __restrict__ a_src,
                           const float* __restrict__ a_dst,
                           float* __restrict__ alS, float* __restrict__ alD) {
  int t = blockIdx.x * blockDim.x + threadIdx.x;  // n*4 + head
  if (t >= NODES * 4) return;
  int n = t >> 2, hd = t & 3;
  const float* hp = H + (size_t)n * DOUT + hd * 16;
  const float* as = a_src + hd * 16;
  const float* ad = a_dst + hd * 16;
  float ss = 0.f, sd = 0.f;
#pragma unroll
  for (int c = 0; c < 16; ++c) {
    float v = hp[c];
    ss = fmaf(v, as[c], ss);
    sd = fmaf(v, ad[c], sd);
  }
  alS[t] = ss;
  alD[t] = sd;
}

// ---- per-edge leaky-relu logit + segment max (keyed atomicMax) ------------
__global__ void edge_max(const int* __restrict__ ei,
                         const float* __restrict__ alS,
                         const float* __restrict__ alD,
                         float* __restrict__ eV, unsigned* __restrict__ mU) {
  int e = blockIdx.x * blockDim.x + threadIdx.x;
  if (e >= TOTE) return;
  int s, d;
  if (e < EDGES) { s = ei[e]; d = ei[EDGES + e]; } else { s = d = e - EDGES; }
#pragma unroll
  for (int hd = 0; hd < 4; ++hd) {
    float v = alS[s * 4 + hd] + alD[d * 4 + hd];
    v = (v > 0.f) ? v : 0.2f * v;   // leaky_relu, slope 0.2
    eV[(size_t)e * 4 + hd] = v;
    atomicMax(&mU[d * 4 + hd], fkey(v));
  }
}

// ---- e = exp(e - max); segment sum --------------------------------------
__global__ void edge_expsum(const int* __restrict__ ei,
                            float* __restrict__ eV,
                            const unsigned* __restrict__ mU,
                            float* __restrict__ sS) {
  int e = blockIdx.x * blockDim.x + threadIdx.x;
  if (e >= TOTE) return;
  int d = (e < EDGES) ? ei[EDGES + e] : (e - EDGES);
#pragma unroll
  for (int hd = 0; hd < 4; ++hd) {
    float m  = funkey(mU[d * 4 + hd]);
    float ex = __expf(eV[(size_t)e * 4 + hd] - m);
    eV[(size_t)e * 4 + hd] = ex;
    atomicAdd(&sS[d * 4 + hd], ex);
  }
}

// ---- aggregate: out[dst] += h[src] * alpha  (one thread per edge-channel) -
__global__ void edge_aggregate(const int* __restrict__ ei,
                               const float* __restrict__ H,
                               const float* __restrict__ eV,
                               const float* __restrict__ sS,
                               float* __restrict__ Out) {
  size_t t = (size_t)blockIdx.x * blockDim.x + threadIdx.x;
  if (t >= (size_t)TOTE * DOUT) return;
  int c = (int)(t & 63);
  int e = (int)(t >> 6);
  int hd = c >> 4;
  int s, d;
  if (e < EDGES) { s = ei[e]; d = ei[EDGES + e]; } else { s = d = e - EDGES; }
  float alpha = eV[(size_t)e * 4 + hd] / (sS[d * 4 + hd] + 1e-16f);
  atomicAdd(&Out[(size_t)d * DOUT + c], H[(size_t)s * DOUT + c] * alpha);
}

// ---- out = relu(out + bias) ----------------------------------------------
__global__ void bias_relu(float* __restrict__ Out, const float* __restrict__ b) {
  size_t i = (size_t)blockIdx.x * blockDim.x + threadIdx.x;
  if (i >= (size_t)NODES * DOUT) return;
  float v = Out[i] + b[i & 63];
  Out[i] = v > 0.f ? v : 0.f;
}

// ---- pooling + linear head ------------------------------------------------
__global__ void pool_init(float* __restrict__ out, const float* __restrict__ head_b) {
  int g = blockIdx.x * blockDim.x + threadIdx.x;
  if (g < GRAPHS) out[g] = head_b[0];
}
__global__ void pool_head(const float* __restrict__ H, const int* __restrict__ batch,
                          const float* __restrict__ head_w, float* __restrict__ out) {
  int n = blockIdx.x * blockDim.x + threadIdx.x;
  if (n >= NODES) return;
  const float* hp = H + (size_t)n * DOUT;
  float acc = 0.f;
#pragma unroll
  for (int c = 0; c < DOUT; ++c) acc = fmaf(hp[c], head_w[c], acc);
  atomicAdd(&out[batch[n]], acc);
}

extern "C" void kernel_launch(void* const* d_in, const int* in_sizes, int n_in,
                              void* d_out, int out_size, void* d_ws, size_t ws_size,
                              hipStream_t stream) {
  const float* x     = (const float*)d_in[0];
  const int*   ei    = (const int*)d_in[1];    // [2,E] int32 (jax default x32)
  const int*   batch = (const int*)d_in[2];
  const float* W[3]  = {(const float*)d_in[3],  (const float*)d_in[7],  (const float*)d_in[11]};
  const float* AS[3] = {(const float*)d_in[4],  (const float*)d_in[8],  (const float*)d_in[12]};
  const float* AD[3] = {(const float*)d_in[5],  (const float*)d_in[9],  (const float*)d_in[13]};
  const float* B[3]  = {(const float*)d_in[6],  (const float*)d_in[10], (const float*)d_in[14]};
  const float* head_w = (const float*)d_in[15];
  const float* head_b = (const float*)d_in[16];
  float* out = (float*)d_out;

  // scratch carve (~85 MB)
  char* p = (char*)d_ws;
  auto carve = [&](size_t bytes) -> void* {
    void* r = (void*)p;
    p += (bytes + 255) & ~(size_t)255;
    return r;
  };
  float*    hA  = (float*)carve(sizeof(float) * (size_t)NODES * DOUT);
  float*    hB  = (float*)carve(sizeof(float) * (size_t)NODES * DOUT);
  float*    eV  = (float*)carve(sizeof(float) * (size_t)TOTE * 4);
  unsigned* mU  = (unsigned*)carve(sizeof(unsigned) * (size_t)NODES * 4);
  float*    sS  = (float*)carve(sizeof(float) * (size_t)NODES * 4);
  float*    alS = (float*)carve(sizeof(float) * (size_t)NODES * 4);
  float*    alD = (float*)carve(sizeof(float) * (size_t)NODES * 4);

  const int tilesPerBlock = 8;                           // 8 waves/block
  const int gemmBlocks = (NODES / 16 + tilesPerBlock - 1) / tilesPerBlock;  // 782
  const int edgeBlocks = (TOTE + 255) / 256;
  const int aggBlocks  = (int)(((size_t)TOTE * DOUT + 255) / 256);
  const int attBlocks  = (NODES * 4 + 255) / 256;
  const int brBlocks   = (int)(((size_t)NODES * DOUT + 255) / 256);

  const float* cur = x;
  for (int l = 0; l < 3; ++l) {
    // feature transform on the matrix pipe (f32 WMMA, exact precision)
    if (l == 0)
      gat_gemm_wmma<128><<<gemmBlocks, 256, 0, stream>>>(cur, W[l], hA);
    else
      gat_gemm_wmma<64><<<gemmBlocks, 256, 0, stream>>>(cur, W[l], hA);

    att_logits<<<attBlocks, 256, 0, stream>>>(hA, AS[l], AD[l], alS, alD);

    hipMemsetAsync(mU, 0, sizeof(unsigned) * (size_t)NODES * 4, stream); // key(any finite)>0
    hipMemsetAsync(sS, 0, sizeof(float) * (size_t)NODES * 4, stream);
    hipMemsetAsync(hB, 0, sizeof(float) * (size_t)NODES * DOUT, stream); // GEMM already consumed hB

    edge_max<<<edgeBlocks, 256, 0, stream>>>(ei, alS, alD, eV, mU);
    edge_expsum<<<edgeBlocks, 256, 0, stream>>>(ei, eV, mU, sS);
    edge_aggregate<<<aggBlocks, 256, 0, stream>>>(ei, hA, eV, sS, hB);
    bias_relu<<<brBlocks, 256, 0, stream>>>(hB, B[l]);
    cur = hB;
  }

  pool_init<<<(GRAPHS + 255) / 256, 256, 0, stream>>>(out, head_b);
  pool_head<<<(NODES + 255) / 256, 256, 0, stream>>>(hB, batch, head_w, out);
}